// _ProposalLayer_8186207666634
// MI455X (gfx1250) — compile-verified
//
#include <hip/hip_runtime.h>
#include <hip/hip_bf16.h>
#include <math.h>

// ---------------- problem constants (from reference) ----------------
#define BB   16          // batch
#define AA   9           // anchors per cell
#define HH   128
#define WW   128
#define NN   (AA*HH*WW)  // 147456 proposals per image
#define PRE  12000       // RPN_PRE_NMS_TOP_N
#define POST 2000        // RPN_POST_NMS_TOP_N
#define FEATS 16.0f
#define MINSZ 8.0f
#define NMST  0.7f
#define NEGF  (-1e30f)

typedef float v2f __attribute__((ext_vector_type(2)));
typedef float v8f __attribute__((ext_vector_type(8)));

// monotonic (order-preserving, descending-friendly) float->u32 key
__device__ __forceinline__ unsigned fkey(float f) {
  unsigned u = __float_as_uint(f);
  return (u & 0x80000000u) ? ~u : (u | 0x80000000u);
}

// ---------------- kernel 0: zero per-image counters ----------------
__global__ void k_init(unsigned* cntG, unsigned* cntE) {
  int t = blockIdx.x * blockDim.x + threadIdx.x;
  if (t < BB) { cntG[t] = 0u; cntE[t] = 0u; }
}

// ---------------- kernel 1: box decode + clip + filter ----------------
__global__ void k_proposals(const float* __restrict__ probs,
                            const float* __restrict__ bbox,
                            const float* __restrict__ imsh,
                            const float* __restrict__ anchors,
                            float* __restrict__ boxes,
                            float* __restrict__ scores,
                            unsigned* __restrict__ keys) {
  int t = blockIdx.x * blockDim.x + threadIdx.x;
  if (t >= BB * NN) return;
  int b   = t / NN;
  int idx = t - b * NN;              // idx = (h*W + w)*A + a
  int a   = idx % AA;
  int hw  = idx / AA;
  int w   = hw % WW;
  int h   = hw / WW;
  const int HW = HH * WW;

  float score = probs[((size_t)(b * (2*AA) + AA + a)) * HW + h * WW + w];
  const float* dp = bbox + ((size_t)(b * (4*AA) + a * 4)) * HW + h * WW + w;
  float dx = dp[0], dy = dp[HW], dw = dp[2*HW], dh = dp[3*HW];

  float sx = (float)w * FEATS, sy = (float)h * FEATS;
  float ax1 = anchors[a*4+0] + sx, ay1 = anchors[a*4+1] + sy;
  float ax2 = anchors[a*4+2] + sx, ay2 = anchors[a*4+3] + sy;

  float wa = ax2 - ax1 + 1.0f, ha = ay2 - ay1 + 1.0f;
  float cx = ax1 + 0.5f * wa,  cy = ay1 + 0.5f * ha;       // matches reference exactly
  float pcx = dx * wa + cx,    pcy = dy * ha + cy;
  float pw  = expf(dw) * wa,   ph  = expf(dh) * ha;

  float imh = imsh[b*2+0] - 1.0f, imw = imsh[b*2+1] - 1.0f;
  float x1 = fminf(fmaxf(pcx - 0.5f*pw, 0.0f), imw);
  float y1 = fminf(fmaxf(pcy - 0.5f*ph, 0.0f), imh);
  float x2 = fminf(fmaxf(pcx + 0.5f*pw, 0.0f), imw);
  float y2 = fminf(fmaxf(pcy + 0.5f*ph, 0.0f), imh);

  float kw = x2 - x1 + 1.0f, kh = y2 - y1 + 1.0f;
  float s = ((kw >= MINSZ) && (kh >= MINSZ)) ? score : NEGF;

  float4 bx = make_float4(x1, y1, x2, y2);
  ((float4*)boxes)[t] = bx;
  scores[t] = s;
  keys[t]   = fkey(s);
}

// ---------------- kernel 2: per-image radix select (threshold for top-12000) ---
// 4 passes over 8-bit digits, MSB first. 256-bin inclusive prefix scan done with
// V_WMMA_F32_16X16X4_F32 on wave 0 (tensor-core scan: S=Lstrict*X, O=S*J, P=X*U+O).
__global__ __launch_bounds__(1024) void k_select(const unsigned* __restrict__ keys,
                                                 unsigned* __restrict__ tkey,
                                                 int* __restrict__ takeEq) {
  __shared__ unsigned histU[256];
  __shared__ float    xF[256];    // X matrix (16x16 row-major), histogram counts
  __shared__ float    sS[256];    // S = Lstrict * X
  __shared__ float    pF[256];    // P = inclusive prefix
  __shared__ unsigned sPrefix;
  __shared__ int      sK;

  const int b   = blockIdx.x;
  const int tid = threadIdx.x;
  if (tid == 0) { sPrefix = 0u; sK = PRE; }
  __syncthreads();

  for (int pass = 0; pass < 4; ++pass) {
    const int shift = 24 - 8 * pass;
    const unsigned maskHi = pass ? (0xFFFFFFFFu << (32 - 8 * pass)) : 0u;

    if (tid < 256) histU[tid] = 0u;
    __syncthreads();
    const unsigned pref = sPrefix;

    for (int i = tid; i < NN; i += 1024) {
      __builtin_prefetch(&keys[(size_t)b * NN + i + 2048], 0, 0); // global_prefetch_b8
      unsigned key = keys[(size_t)b * NN + i];
      if ((key & maskHi) == pref)
        atomicAdd(&histU[(key >> shift) & 255u], 1u);
    }
    __syncthreads();
    if (tid < 256) xF[tid] = (float)histU[tid];   // exact: counts < 2^24
    __syncthreads();

    // ---- WMMA prefix scan over 256 bins (wave 0, EXEC all-1s) ----
    if (tid < 32) {
      const int lane = tid;
      const int half = lane >> 4;     // 0: K pair {0,1}; 1: K pair {2,3}
      const int m16  = lane & 15;     // A-row / B-col / D-col index

      // S = Lstrict * X   (S[i][j] = sum_{k<i} X[k][j])
      v8f dS = {};
      for (int t4 = 0; t4 < 4; ++t4) {
        int k0 = 4 * t4 + half * 2;
        v2f av, bv;
        av.x = (float)((k0 + 0) < m16);            // Lstrict[M=m16][K]
        av.y = (float)((k0 + 1) < m16);
        bv.x = xF[(k0 + 0) * 16 + m16];            // X[K][N=m16]
        bv.y = xF[(k0 + 1) * 16 + m16];
        dS = __builtin_amdgcn_wmma_f32_16x16x4_f32(false, av, false, bv,
                                                   (short)0, dS, false, false);
      }
      for (int q = 0; q < 8; ++q) sS[(q + half * 8) * 16 + m16] = dS[q];

      // O = S * J   (O[i][j] = sum of all rows < i of X)
      v8f dO = {};
      for (int t4 = 0; t4 < 4; ++t4) {
        int k0 = 4 * t4 + half * 2;
        v2f av, bv;
        av.x = sS[m16 * 16 + k0 + 0];              // S[M=m16][K]
        av.y = sS[m16 * 16 + k0 + 1];
        bv.x = 1.0f; bv.y = 1.0f;                  // J (all ones)
        dO = __builtin_amdgcn_wmma_f32_16x16x4_f32(false, av, false, bv,
                                                   (short)0, dO, false, false);
      }

      // P = X * U_incl + O  (inclusive scan over linear index m = 16*i + j)
      v8f dP = dO;
      for (int t4 = 0; t4 < 4; ++t4) {
        int k0 = 4 * t4 + half * 2;
        v2f av, bv;
        av.x = xF[m16 * 16 + k0 + 0];              // X[M=m16][K]
        av.y = xF[m16 * 16 + k0 + 1];
        bv.x = (float)((k0 + 0) <= m16);           // U[K][N=m16]
        bv.y = (float)((k0 + 1) <= m16);
        dP = __builtin_amdgcn_wmma_f32_16x16x4_f32(false, av, false, bv,
                                                   (short)0, dP, false, false);
      }
      for (int q = 0; q < 8; ++q) pF[(q + half * 8) * 16 + m16] = dP[q];
    }
    __syncthreads();

    if (tid == 0) {
      float total = pF[255];
      int k = sK;
      int sel = 255;
      for (int m = 0; m < 256; ++m) {
        if ((int)(total - pF[m]) < k) { sel = m; break; }   // greater(m) < k
      }
      sK = k - (int)(total - pF[sel]);
      sPrefix = pref | ((unsigned)sel << shift);
    }
    __syncthreads();
  }
  if (tid == 0) { tkey[b] = sPrefix; takeEq[b] = sK; }
}

// ---------------- kernel 3: compact exactly PRE candidates per image ----------
__global__ void k_compact(const float* __restrict__ boxes,
                          const float* __restrict__ scores,
                          const unsigned* __restrict__ keys,
                          const unsigned* __restrict__ tkey,
                          const int* __restrict__ takeEq,
                          unsigned* __restrict__ cntG,
                          unsigned* __restrict__ cntE,
                          float* __restrict__ cboxes,
                          float* __restrict__ cscores) {
  int t = blockIdx.x * blockDim.x + threadIdx.x;
  if (t >= BB * NN) return;
  int b = t / NN;
  unsigned key = keys[t];
  unsigned Tk  = tkey[b];
  int slot = -1;
  if (key > Tk) {
    slot = (int)atomicAdd(&cntG[b], 1u);             // < PRE - takeEq by construction
  } else if (key == Tk) {
    unsigned pe = atomicAdd(&cntE[b], 1u);
    int te = takeEq[b];
    if ((int)pe < te) slot = (PRE - te) + (int)pe;
  }
  if (slot >= 0) {
    ((float4*)cboxes)[(size_t)b * PRE + slot] = ((const float4*)boxes)[t];
    cscores[(size_t)b * PRE + slot] = scores[t];
  }
}

// ---------------- kernel 4: greedy NMS, whole working set in 240KB LDS -------
// One 1024-thread workgroup (32 wave32s) per image. CDNA5's 320KB-per-WGP LDS
// holds boxes (AoS float4, 192KB) + scores (48KB). The fill uses the gfx1250
// async copy path (GLOBAL_LOAD_ASYNC_TO_LDS_B128/B32, ASYNCcnt-tracked) so the
// 240KB staging never round-trips through VGPRs.
__global__ __launch_bounds__(1024, 1) void k_nms(const float* __restrict__ cboxes,
                                                 const float* __restrict__ cscores,
                                                 float* __restrict__ out) {
  extern __shared__ float smem[];
  float* sbox = smem;              // [PRE][4]  (AoS, 16B per candidate)
  float* ss   = smem + 4 * PRE;    // [PRE]
  __shared__ float rv[32];
  __shared__ int   ri[32];
  __shared__ int   sj;

  const int b = blockIdx.x, tid = threadIdx.x;
  const int lane = tid & 31, wid = tid >> 5;

  // ---- async fill: global (L2) -> LDS, GVS mode (SGPR base + VGPR offset) ----
  {
    unsigned long long gbox = (unsigned long long)(cboxes + (size_t)b * PRE * 4);
    unsigned long long gsc  = (unsigned long long)(cscores + (size_t)b * PRE);
    unsigned ldsBoxBase = (unsigned)(uintptr_t)sbox;   // LDS aperture: addr[31:0]
    unsigned ldsScBase  = (unsigned)(uintptr_t)ss;
    for (int i = tid; i < PRE; i += 1024) {
      unsigned ldsBox = ldsBoxBase + (unsigned)i * 16u;
      unsigned offBox = (unsigned)i * 16u;
      asm volatile("global_load_async_to_lds_b128 %0, %1, %2"
                   :: "v"(ldsBox), "v"(offBox), "s"(gbox) : "memory");
      unsigned ldsSc = ldsScBase + (unsigned)i * 4u;
      unsigned offSc = (unsigned)i * 4u;
      asm volatile("global_load_async_to_lds_b32 %0, %1, %2"
                   :: "v"(ldsSc), "v"(offSc), "s"(gsc) : "memory");
    }
    asm volatile("s_wait_asynccnt 0x0" ::: "memory");
  }
  __syncthreads();

  for (int it = 0; it < POST; ++it) {
    // ---- argmax over PRE scores (tie: smallest index) ----
    float best = -3.4e38f; int bi = 0x7FFFFFFF;
    for (int i = tid; i < PRE; i += 1024) {
      float v = ss[i];
      if (v > best || (v == best && i < bi)) { best = v; bi = i; }
    }
    for (int off = 16; off; off >>= 1) {
      float ov = __shfl_down(best, off, 32);
      int   oi = __shfl_down(bi,   off, 32);
      if (ov > best || (ov == best && oi < bi)) { best = ov; bi = oi; }
    }
    if (lane == 0) { rv[wid] = best; ri[wid] = bi; }
    __syncthreads();
    if (tid < 32) {
      best = rv[tid]; bi = ri[tid];
      for (int off = 16; off; off >>= 1) {
        float ov = __shfl_down(best, off, 32);
        int   oi = __shfl_down(bi,   off, 32);
        if (ov > best || (ov == best && oi < bi)) { best = ov; bi = oi; }
      }
      if (tid == 0) {
        bool valid = best > (NEGF * 0.5f);
        sj = valid ? bi : -1;
        float* o = out + ((size_t)b * POST + it) * 5;
        if (valid) {
          o[0] = (float)b;
          o[1] = sbox[bi*4+0]; o[2] = sbox[bi*4+1];
          o[3] = sbox[bi*4+2]; o[4] = sbox[bi*4+3];
        } else {
          o[0] = 0.f; o[1] = 0.f; o[2] = 0.f; o[3] = 0.f; o[4] = 0.f;
        }
      }
    }
    __syncthreads();

    int j = sj;
    if (j < 0) {
      // no valid candidates remain: zero-fill remaining rows and stop
      size_t base = ((size_t)b * POST + it + 1) * 5;
      int remain = (POST - 1 - it) * 5;
      for (int q = tid; q < remain; q += 1024) out[base + q] = 0.f;
      break;
    }

    // ---- suppress IoU > thresh against chosen box ----
    float4 jb = ((const float4*)sbox)[j];
    float ja = (jb.z - jb.x + 1.0f) * (jb.w - jb.y + 1.0f);
    for (int i = tid; i < PRE; i += 1024) {
      if (ss[i] <= (NEGF * 0.5f)) continue;          // already dead: skip box load
      float4 ib = ((const float4*)sbox)[i];
      float xx1 = fmaxf(jb.x, ib.x), yy1 = fmaxf(jb.y, ib.y);
      float xx2 = fminf(jb.z, ib.z), yy2 = fminf(jb.w, ib.w);
      float iw = fmaxf(xx2 - xx1 + 1.0f, 0.0f);
      float ih = fmaxf(yy2 - yy1 + 1.0f, 0.0f);
      float inter = iw * ih;
      float ia = (ib.z - ib.x + 1.0f) * (ib.w - ib.y + 1.0f);
      float iou = inter / (ja + ia - inter);
      if (iou > NMST) ss[i] = NEGF;
    }
    __syncthreads();
  }
}

// ---------------- host launcher ----------------
extern "C" void kernel_launch(void* const* d_in, const int* in_sizes, int n_in,
                              void* d_out, int out_size, void* d_ws, size_t ws_size,
                              hipStream_t stream) {
  const float* probs   = (const float*)d_in[0];   // (B, 2A, H, W)
  const float* bbox    = (const float*)d_in[1];   // (B, 4A, H, W)
  const float* imsh    = (const float*)d_in[2];   // (B, 2)
  const float* anchors = (const float*)d_in[3];   // (9, 4)
  float* out = (float*)d_out;                     // (B, 2000, 5)

  // workspace layout (all 16B aligned)
  char* ws = (char*)d_ws;
  float*    boxes   = (float*)   (ws);                               // B*N*4 f32
  float*    scores  = (float*)   (ws + 37748736);                    // B*N   f32
  unsigned* keys    = (unsigned*)(ws + 47185920);                    // B*N   u32
  float*    cboxes  = (float*)   (ws + 56623104);                    // B*PRE*4
  float*    cscores = (float*)   (ws + 59695104);                    // B*PRE
  unsigned* tkey    = (unsigned*)(ws + 60463104);                    // B
  int*      takeEq  = (int*)     (ws + 60463104 + 64);               // B
  unsigned* cntG    = (unsigned*)(ws + 60463104 + 128);              // B
  unsigned* cntE    = (unsigned*)(ws + 60463104 + 192);              // B

  const int total = BB * NN;
  k_init<<<1, 64, 0, stream>>>(cntG, cntE);
  k_proposals<<<(total + 255) / 256, 256, 0, stream>>>(probs, bbox, imsh, anchors,
                                                       boxes, scores, keys);
  k_select<<<BB, 1024, 0, stream>>>(keys, tkey, takeEq);
  k_compact<<<(total + 255) / 256, 256, 0, stream>>>(boxes, scores, keys, tkey, takeEq,
                                                     cntG, cntE, cboxes, cscores);
  k_nms<<<BB, 1024, (size_t)5 * PRE * sizeof(float), stream>>>(cboxes, cscores, out);
}